// TFFirUpsample2D_44186623541793
// MI455X (gfx1250) — compile-verified
//
#include <hip/hip_runtime.h>

// ---------------------------------------------------------------------------
// TF FIR upsample2d (up=2, k=[1,3,3,1] outer, gain*factor^2) on MI455X.
//
// Bandwidth-bound: 67MB read + 268MB write @ 23.3TB/s ~= 14.4us floor;
// ~1.1 GFLOP of FMAs is noise. CDNA5 path used: async global->LDS B128
// staging (ASYNCcnt), wave32 128-bit channel vectors, NT stores.
// ---------------------------------------------------------------------------

typedef float v4f __attribute__((ext_vector_type(4)));

#define N_      8
#define H_      64
#define W_      64
#define C_      512
#define OH_     (2 * H_)
#define OW_     (2 * W_)

#define CBLK    32              // channels per block
#define CVEC    (CBLK / 4)      // 8 float4 lanes in C
#define NCB     (C_ / CBLK)     // 16 channel slabs
#define RSTRIP  4               // input rows produced per block (-> 8 output rows)
#define NSTRIP  (H_ / RSTRIP)   // 16 row strips
#define LROWS   (RSTRIP + 2)    // staged rows incl. +/-1 halo

#define THREADS 256

__global__ __launch_bounds__(THREADS)
void fir_up2_kernel(const float* __restrict__ x, float* __restrict__ out) {
    // 6 rows x 64 cols x 8 float4 = 48 KB of LDS
    __shared__ v4f tile[LROWS][W_][CVEC];

    const int tid = threadIdx.x;
    int b = blockIdx.x;
    const int cb = b % NCB;   b /= NCB;
    const int ys = b % NSTRIP; b /= NSTRIP;
    const int n  = b;
    const int iy0 = ys * RSTRIP;

    // -------------------------------------------------------------------
    // Stage input rows [iy0-1, iy0+RSTRIP] (row-clamped; invalid rows are
    // zeroed at compute time) into LDS with async B128 DMA (no VGPR hop).
    // GVS addressing: SGPR 64-bit base + per-lane 32-bit byte offset.
    // -------------------------------------------------------------------
    const uint64_t sbase =
        (uint64_t)(uintptr_t)(x + ((size_t)n * H_ * W_) * C_ + (size_t)cb * CBLK);
    // low 32 bits of the generic address of a __shared__ object are the LDS
    // byte offset (shared aperture lives in the high bits).
    const uint32_t lds_base = (uint32_t)(uintptr_t)(&tile[0][0][0]);

    #pragma unroll
    for (int t = 0; t < (LROWS * W_ * CVEC) / THREADS; ++t) {   // 12 iterations
        const int l   = tid + THREADS * t;                      // 0..3071
        const int cv  = l & (CVEC - 1);
        const int col = (l >> 3) & (W_ - 1);
        int r         = l >> 9;                                 // 0..5 (64*8=512/row)
        int gy        = iy0 - 1 + r;
        gy = gy < 0 ? 0 : (gy > H_ - 1 ? H_ - 1 : gy);          // clamp (halo)
        const uint32_t voff =
            (uint32_t)((((gy * W_ + col) * C_) + cv * 4) * sizeof(float));
        const uint32_t ldsa =
            lds_base + (uint32_t)((((r * W_) + col) * CVEC + cv) * sizeof(v4f));
        asm volatile("global_load_async_to_lds_b128 %0, %1, %2"
                     :
                     : "v"(ldsa), "v"(voff), "s"(sbase)
                     : "memory");
    }
    asm volatile("s_wait_asynccnt 0" ::: "memory");
    __syncthreads();

    // -------------------------------------------------------------------
    // Compute. Each input pixel (iy, j) yields a 2x2 output quad:
    //   even tap pair {0.25,0.75} over (prev,cur), odd {0.75,0.25} over
    //   (cur,next), separably in x and y. Horizontal blends L/R per staged
    //   row are rolled vertically through the strip in registers.
    // -------------------------------------------------------------------
    const int cv = tid & (CVEC - 1);
    const int jg = tid >> 3;                  // 0..31
    const bool top_edge = (iy0 == 0);
    const bool bot_edge = (iy0 + RSTRIP == H_);
    const v4f z = {0.0f, 0.0f, 0.0f, 0.0f};

    float* const obase =
        out + ((size_t)n * OH_ * OW_) * C_ + (size_t)cb * CBLK + cv * 4;

    auto row_blend = [&](int rI, int j, bool valid, v4f& L, v4f& R) {
        if (!valid) { L = z; R = z; return; }
        const v4f a = (j > 0)      ? tile[rI][j - 1][cv] : z;
        const v4f c0 =               tile[rI][j    ][cv];
        const v4f c1 = (j < W_ - 1) ? tile[rI][j + 1][cv] : z;
        L = 0.25f * a  + 0.75f * c0;   // ox = 2j
        R = 0.75f * c0 + 0.25f * c1;   // ox = 2j + 1
    };

    #pragma unroll
    for (int k = 0; k < W_ / 32; ++k) {       // 2 column groups
        const int j = jg + 32 * k;            // 0..63

        v4f L0, R0, L1, R1, L2, R2;
        row_blend(0, j, !top_edge, L0, R0);   // gy = iy0-1
        row_blend(1, j, true,      L1, R1);   // gy = iy0

        #pragma unroll
        for (int ly = 0; ly < RSTRIP; ++ly) {
            const int  rI = ly + 2;                        // gy = iy0+ly+1
            const bool v  = !(bot_edge && rI == LROWS - 1);
            row_blend(rI, j, v, L2, R2);

            const v4f o00 = 0.25f * L0 + 0.75f * L1;       // (2iy,   2j)
            const v4f o01 = 0.25f * R0 + 0.75f * R1;       // (2iy,   2j+1)
            const v4f o10 = 0.75f * L1 + 0.25f * L2;       // (2iy+1, 2j)
            const v4f o11 = 0.75f * R1 + 0.25f * R2;       // (2iy+1, 2j+1)

            const int oy = 2 * (iy0 + ly);
            float* const p0 = obase + ((size_t)oy * OW_ + 2 * j) * C_;
            float* const p1 = p0 + (size_t)OW_ * C_;
            __builtin_nontemporal_store(o00, (v4f*)(p0));
            __builtin_nontemporal_store(o01, (v4f*)(p0 + C_));
            __builtin_nontemporal_store(o10, (v4f*)(p1));
            __builtin_nontemporal_store(o11, (v4f*)(p1 + C_));

            L0 = L1; R0 = R1; L1 = L2; R1 = R2;
        }
    }
}

extern "C" void kernel_launch(void* const* d_in, const int* in_sizes, int n_in,
                              void* d_out, int out_size, void* d_ws, size_t ws_size,
                              hipStream_t stream) {
    const float* x = (const float*)d_in[0];
    float* out     = (float*)d_out;

    const dim3 grid(N_ * NSTRIP * NCB);   // 8 * 16 * 16 = 2048 blocks
    const dim3 block(THREADS);            // 8 wave32
    hipLaunchKernelGGL(fir_up2_kernel, grid, block, 0, stream, x, out);
}